// GNN_7224134992541
// MI455X (gfx1250) — compile-verified
//
#include <hip/hip_runtime.h>
#include <math.h>

#define N_NODES 50000
#define N_EDGES 800000
#define FDIM    128
#define NOUT    10
#define NGRAPH  64

typedef __attribute__((ext_vector_type(2))) float v2f;
typedef __attribute__((ext_vector_type(8))) float v8f;

// ---------------- degree / norm ----------------------------------------

__global__ void init_deg(float* deg) {
    int i = blockIdx.x * blockDim.x + threadIdx.x;
    if (i < N_NODES) deg[i] = 1.0f;            // self-loop weight
}

__global__ void accum_deg(const int* __restrict__ dst,
                          const float* __restrict__ w, float* deg) {
    int e = blockIdx.x * blockDim.x + threadIdx.x;
    if (e < N_EDGES) atomicAdd(&deg[dst[e]], w[e]);
}

__global__ void finish_dinv(float* dinv) {
    int i = blockIdx.x * blockDim.x + threadIdx.x;
    if (i < N_NODES) {
        float d = dinv[i];
        dinv[i] = (d > 0.0f) ? rsqrtf(d) : 0.0f;
    }
}

__global__ void compute_norm(const int* __restrict__ src,
                             const int* __restrict__ dst,
                             const float* __restrict__ w,
                             const float* __restrict__ dinv,
                             float* __restrict__ norm) {
    int e = blockIdx.x * blockDim.x + threadIdx.x;
    if (e < N_EDGES) norm[e] = dinv[src[e]] * w[e] * dinv[dst[e]];
}

// ---------------- dense transform: H = (relu?)X @ W, f32 WMMA ----------
// grid.x = N/16 row tiles; 256 threads = 8 waves; wave w owns col tile w.
// ReLU is a compile-time parameter: layer 1 gets a clean load->wmma chain,
// layer 2 gets a single v_max_num per element.

template <bool APPLY_RELU>
__global__ __launch_bounds__(256) void gemm128_wmma(
        const float* __restrict__ X, const float* __restrict__ W,
        float* __restrict__ H) {
    const int lane = threadIdx.x & 31;
    const int wid  = threadIdx.x >> 5;
    const int row0 = blockIdx.x << 4;
    const int l15  = lane & 15;
    const int koff = (lane >> 4) << 1;      // lanes 0-15 -> K 0,1 ; 16-31 -> K 2,3
    const float* arow = X + (size_t)(row0 + l15) * FDIM + koff;
    const float* bcol = W + (wid << 4) + l15;
    v8f acc = {};
#pragma unroll
    for (int k = 0; k < FDIM; k += 4) {
        v2f a, b;
        float x0 = arow[k];
        float x1 = arow[k + 1];
        if (APPLY_RELU) { x0 = fmaxf(x0, 0.0f); x1 = fmaxf(x1, 0.0f); }
        a.x = x0; a.y = x1;
        b.x = bcol[(size_t)(k + koff) * FDIM];
        b.y = bcol[(size_t)(k + koff + 1) * FDIM];
        acc = __builtin_amdgcn_wmma_f32_16x16x4_f32(
                  false, a, false, b, (short)0, acc, false, false);
    }
    const int ncol = (wid << 4) + l15;
    const int rb   = row0 + ((lane >> 4) << 3);   // VGPR i -> row rb+i
    float* out = H + (size_t)rb * FDIM + ncol;
#pragma unroll
    for (int i = 0; i < 8; ++i) out[(size_t)i * FDIM] = acc[i];
}

// ---------------- aggregation: agg = b + dinv^2*h + scatter ------------

__global__ void agg_init128(const float* __restrict__ h,
                            const float* __restrict__ dinv,
                            const float* __restrict__ bias,
                            float* __restrict__ agg) {
    size_t idx = (size_t)blockIdx.x * blockDim.x + threadIdx.x;
    if (idx < (size_t)N_NODES * FDIM) {
        int n = (int)(idx >> 7);
        int f = (int)(idx & 127);
        float di = dinv[n];
        agg[idx] = bias[f] + di * di * h[idx];
    }
}

// one wave per edge; lane handles 4 consecutive features
__global__ __launch_bounds__(256) void agg_scatter128(
        const float* __restrict__ h, const float* __restrict__ norm,
        const int* __restrict__ src, const int* __restrict__ dst,
        float* __restrict__ agg) {
    int e = blockIdx.x * 8 + (threadIdx.x >> 5);
    if (e >= N_EDGES) return;
    int lane = threadIdx.x & 31;
    int s = src[e], d = dst[e];
    float w = norm[e];
    const float4* hv = (const float4*)(h + (size_t)s * FDIM);
    float4 v = hv[lane];
    float* o = agg + (size_t)d * FDIM + lane * 4;
    atomicAdd(o + 0, w * v.x);
    atomicAdd(o + 1, w * v.y);
    atomicAdd(o + 2, w * v.z);
    atomicAdd(o + 3, w * v.w);
}

// ---------------- head: h3 = relu(agg2) @ W3 (128 -> 10) ---------------

__global__ void lin10(const float* __restrict__ A, const float* __restrict__ W3,
                      float* __restrict__ H3) {
    int idx = blockIdx.x * blockDim.x + threadIdx.x;
    if (idx >= N_NODES * NOUT) return;
    int n = idx / NOUT, o = idx % NOUT;
    const float* a = A + (size_t)n * FDIM;
    float s = 0.0f;
#pragma unroll 8
    for (int k = 0; k < FDIM; ++k) s += fmaxf(a[k], 0.0f) * W3[k * NOUT + o];
    H3[idx] = s;
}

__global__ void agg_init10(const float* __restrict__ h3,
                           const float* __restrict__ dinv,
                           const float* __restrict__ b3,
                           float* __restrict__ o3) {
    int idx = blockIdx.x * blockDim.x + threadIdx.x;
    if (idx >= N_NODES * NOUT) return;
    int n = idx / NOUT, o = idx % NOUT;
    float di = dinv[n];
    o3[idx] = b3[o] + di * di * h3[idx];
}

__global__ void agg_scatter10(const float* __restrict__ h3,
                              const float* __restrict__ norm,
                              const int* __restrict__ src,
                              const int* __restrict__ dst,
                              float* __restrict__ o3) {
    int e = blockIdx.x * blockDim.x + threadIdx.x;
    if (e >= N_EDGES) return;
    int s = src[e], d = dst[e];
    float w = norm[e];
    const float* hs = h3 + (size_t)s * NOUT;
    float* od = o3 + (size_t)d * NOUT;
#pragma unroll
    for (int o = 0; o < NOUT; ++o) atomicAdd(&od[o], w * hs[o]);
}

// ---------------- pooling + log_softmax --------------------------------

__global__ void pool_zero(float* sums, float* cnts) {
    int i = blockIdx.x * blockDim.x + threadIdx.x;
    if (i < NGRAPH * NOUT) sums[i] = 0.0f;
    if (i < NGRAPH) cnts[i] = 0.0f;
}

__global__ void pool_accum(const float* __restrict__ o3,
                           const int* __restrict__ batch,
                           float* sums, float* cnts) {
    int n = blockIdx.x * blockDim.x + threadIdx.x;
    if (n >= N_NODES) return;
    int g = batch[n];
    atomicAdd(&cnts[g], 1.0f);
    const float* r = o3 + (size_t)n * NOUT;
#pragma unroll
    for (int o = 0; o < NOUT; ++o) atomicAdd(&sums[g * NOUT + o], r[o]);
}

__global__ void final_lsm(const float* __restrict__ sums,
                          const float* __restrict__ cnts,
                          float* __restrict__ out) {
    int g = threadIdx.x;
    if (g >= NGRAPH) return;
    float c = fmaxf(cnts[g], 1.0f);
    float v[NOUT];
    float m = -3.402823e38f;
#pragma unroll
    for (int o = 0; o < NOUT; ++o) { v[o] = sums[g * NOUT + o] / c; m = fmaxf(m, v[o]); }
    float s = 0.0f;
#pragma unroll
    for (int o = 0; o < NOUT; ++o) s += expf(v[o] - m);
    float lse = m + logf(s);
#pragma unroll
    for (int o = 0; o < NOUT; ++o) out[g * NOUT + o] = v[o] - lse;
}

// ---------------- host driver ------------------------------------------

extern "C" void kernel_launch(void* const* d_in, const int* in_sizes, int n_in,
                              void* d_out, int out_size, void* d_ws, size_t ws_size,
                              hipStream_t stream) {
    const float* x    = (const float*)d_in[0];
    const int*   ei   = (const int*)d_in[1];
    const float* ew   = (const float*)d_in[2];
    const int*   batch= (const int*)d_in[3];
    const float* W1   = (const float*)d_in[4];
    const float* b1   = (const float*)d_in[5];
    const float* W2   = (const float*)d_in[6];
    const float* b2   = (const float*)d_in[7];
    const float* W3   = (const float*)d_in[8];
    const float* b3   = (const float*)d_in[9];
    float* out = (float*)d_out;

    char* base = (char*)d_ws;
    size_t cur = 0;
    auto alloc = [&](size_t bytes) -> float* {
        float* p = (float*)(base + cur);
        cur = (cur + bytes + 255) & ~(size_t)255;
        return p;
    };
    float* dinv = alloc((size_t)N_NODES * 4);
    float* norm = alloc((size_t)N_EDGES * 4);
    float* bufA = alloc((size_t)N_NODES * FDIM * 4);
    float* bufB = alloc((size_t)N_NODES * FDIM * 4);
    float* h3   = alloc((size_t)N_NODES * NOUT * 4);
    float* o3   = alloc((size_t)N_NODES * NOUT * 4);
    float* sums = alloc((size_t)NGRAPH * NOUT * 4);
    float* cnts = alloc((size_t)NGRAPH * 4);

    const int* src = ei;
    const int* dst = ei + N_EDGES;

    const int TB = 256;
    int nbN   = (N_NODES + TB - 1) / TB;
    int nbE   = (N_EDGES + TB - 1) / TB;
    int nbNF  = (int)(((size_t)N_NODES * FDIM + TB - 1) / TB);
    int nbN10 = (N_NODES * NOUT + TB - 1) / TB;
    int nbEW  = (N_EDGES + 7) / 8;        // 8 edge-waves per 256-thread block
    int nbRT  = N_NODES / 16;             // 3125 exact row tiles

    // normalization
    init_deg<<<nbN, TB, 0, stream>>>(dinv);
    accum_deg<<<nbE, TB, 0, stream>>>(dst, ew, dinv);
    finish_dinv<<<nbN, TB, 0, stream>>>(dinv);
    compute_norm<<<nbE, TB, 0, stream>>>(src, dst, ew, dinv, norm);

    // layer 1: h = x@W1 ; agg = b1 + dinv^2 h + scatter
    gemm128_wmma<false><<<nbRT, TB, 0, stream>>>(x, W1, bufA);
    agg_init128<<<nbNF, TB, 0, stream>>>(bufA, dinv, b1, bufB);
    agg_scatter128<<<nbEW, TB, 0, stream>>>(bufA, norm, src, dst, bufB);

    // layer 2: h = relu(agg1)@W2 ; agg = b2 + dinv^2 h + scatter
    gemm128_wmma<true><<<nbRT, TB, 0, stream>>>(bufB, W2, bufA);
    agg_init128<<<nbNF, TB, 0, stream>>>(bufA, dinv, b2, bufB);
    agg_scatter128<<<nbEW, TB, 0, stream>>>(bufA, norm, src, dst, bufB);

    // layer 3: h3 = relu(agg2)@W3 ; o3 = b3 + dinv^2 h3 + scatter
    lin10<<<nbN10, TB, 0, stream>>>(bufB, W3, h3);
    agg_init10<<<nbN10, TB, 0, stream>>>(h3, dinv, b3, o3);
    agg_scatter10<<<nbE, TB, 0, stream>>>(h3, norm, src, dst, o3);

    // global mean pool + log_softmax
    pool_zero<<<3, TB, 0, stream>>>(sums, cnts);
    pool_accum<<<nbN, TB, 0, stream>>>(o3, batch, sums, cnts);
    final_lsm<<<1, 64, 0, stream>>>(sums, cnts, out);
}